// ScaledDotProductAttention_68599217652302
// MI455X (gfx1250) — compile-verified
//
#include <hip/hip_runtime.h>

typedef __attribute__((ext_vector_type(2))) float v2f;
typedef __attribute__((ext_vector_type(4))) float v4f;
typedef __attribute__((ext_vector_type(8))) float v8f;

#define BQ   64
#define SQ   1024
#define DH   64
#define NEGV (-1e9f)

// One workgroup = 4 waves (128 threads, wave32) handles 16 query rows of one batch.
// LDS holds the 16 x 1024 f32 logit/prob tile (64 KB).
__global__ __launch_bounds__(128) void attn_fused_kernel(
    const float* __restrict__ q,
    const float* __restrict__ k,
    const float* __restrict__ v,
    const unsigned char* __restrict__ mask,   // causal; recomputed from indices
    const float* __restrict__ prior,
    float* __restrict__ out,                  // [B,S,D]
    float* __restrict__ attn_out)             // [B,S,S]
{
    __shared__ float lds[16 * SQ];

    const int b      = blockIdx.y;
    const int qtile  = blockIdx.x;
    const int qbase  = qtile * 16;
    const int tid    = threadIdx.x;
    const int wave   = tid >> 5;
    const int lane   = tid & 31;
    const int lhalf  = lane >> 4;   // 0: lanes 0-15, 1: lanes 16-31
    const int l16    = lane & 15;
    const int nlimit = (qtile + 1) * 16;   // causal active key range

    // ---- Preload Q A-fragments (16 steps of K=4) -------------------------
    // A 16x4 f32 layout: lanes 0-15 hold (M=lane, K={d0,d0+1}), lanes 16-31
    // hold (M=lane-16, K={d0+2,d0+3}).
    v2f qa[16];
    {
        const float* qrow = q + ((size_t)b * SQ + qbase + l16) * DH + lhalf * 2;
#pragma unroll
        for (int s = 0; s < 16; ++s)
            qa[s] = *(const v2f*)(qrow + 4 * s);
    }

    // ---- Phase A: logits = gate(QK^T/8, prior) into LDS ------------------
    for (int t = wave; t <= qtile; t += 4) {
        const int nbase = t * 16;
        const float* krow = k + ((size_t)b * SQ + nbase + l16) * DH + lhalf * 2;
        v8f acc = {};
#pragma unroll
        for (int s = 0; s < 16; ++s) {
            v2f kb = *(const v2f*)(krow + 4 * s);
            acc = __builtin_amdgcn_wmma_f32_16x16x4_f32(
                false, qa[s], false, kb, (short)0, acc, false, false);
        }
        // C/D layout: VGPR r -> (M=r, N=l16) for lanes 0-15, (M=r+8) for 16-31
        const int n_g = nbase + l16;
#pragma unroll
        for (int r = 0; r < 8; ++r) {
            const int m_loc = r + lhalf * 8;
            const int m_g   = qbase + m_loc;
            float a = acc[r] * 0.125f;                       // 1/sqrt(64)
            // prior is streamed exactly once -> non-temporal load (TH=NT)
            float p = __builtin_nontemporal_load(
                prior + ((size_t)b * SQ + m_g) * SQ + n_g);
            if (n_g > m_g) { a = NEGV; p = NEGV; }           // causal mask
            float z = 1.0f / (1.0f + __expf(-(a + p)));      // sigmoid gate
            lds[m_loc * SQ + n_g] = z * a + (1.0f - z) * p;
        }
    }
    __syncthreads();

    // ---- Phase B: row softmax; write probs to LDS + global ---------------
#pragma unroll
    for (int rr = 0; rr < 4; ++rr) {
        const int m_loc = wave * 4 + rr;
        const int m_g   = qbase + m_loc;
        float* row = lds + m_loc * SQ;

        float mx = -3.4e38f;
        for (int c = lane; c < nlimit; c += 32) mx = fmaxf(mx, row[c]);
#pragma unroll
        for (int off = 16; off >= 1; off >>= 1)
            mx = fmaxf(mx, __shfl_xor(mx, off, 32));

        float sum = 0.0f;
        for (int c = lane; c < nlimit; c += 32) {
            float e = __expf(row[c] - mx);
            row[c] = e;
            sum += e;
        }
#pragma unroll
        for (int off = 16; off >= 1; off >>= 1)
            sum += __shfl_xor(sum, off, 32);
        const float inv = 1.0f / sum;

        // attn output is write-once streaming data: non-temporal stores keep
        // the 268 MB stream from evicting K/V/Q from the 192 MB L2.
        float* gr = attn_out + ((size_t)b * SQ + m_g) * SQ;
        for (int c = 4 * lane; c < nlimit; c += 128) {
            v4f pv = *(v4f*)(row + c);
            pv *= inv;
            *(v4f*)(row + c) = pv;                      // keep probs for PV matmul
            __builtin_nontemporal_store(pv, (v4f*)(gr + c));
        }
        const v4f zz = {};
        for (int c = nlimit + 4 * lane; c < SQ; c += 128)
            __builtin_nontemporal_store(zz, (v4f*)(gr + c));  // masked region -> 0
    }
    __syncthreads();

    // ---- Phase C: out = attn @ V (wave owns D-tile dbase..dbase+15) ------
    {
        const int dbase = wave * 16;
        v8f acc = {};
        const float* lrow  = lds + l16 * SQ + lhalf * 2;
        const float* vbase = v + (size_t)b * SQ * DH + (size_t)(lhalf * 2) * DH
                               + dbase + l16;
        for (int kk = 0; kk < nlimit; kk += 4) {
            v2f a = *(const v2f*)(lrow + kk);          // A from LDS probs
            const float* vp = vbase + (size_t)kk * DH; // B[K][N] = v[kk+K][dbase+N]
            v2f bf;
            bf.x = vp[0];
            bf.y = vp[DH];
            acc = __builtin_amdgcn_wmma_f32_16x16x4_f32(
                false, a, false, bf, (short)0, acc, false, false);
        }
#pragma unroll
        for (int r = 0; r < 8; ++r) {
            const int m_loc = r + lhalf * 8;
            out[((size_t)b * SQ + qbase + m_loc) * DH + dbase + l16] = acc[r];
        }
    }
    (void)mask;
}

extern "C" void kernel_launch(void* const* d_in, const int* in_sizes, int n_in,
                              void* d_out, int out_size, void* d_ws, size_t ws_size,
                              hipStream_t stream) {
    const float*         q     = (const float*)d_in[0];
    const float*         k     = (const float*)d_in[1];
    const float*         v     = (const float*)d_in[2];
    const unsigned char* mask  = (const unsigned char*)d_in[3];
    const float*         prior = (const float*)d_in[4];

    float* out  = (float*)d_out;                       // [B,S,D] first
    float* attn = out + (size_t)BQ * SQ * DH;          // then [B,S,S]

    dim3 grid(SQ / 16, BQ);
    dim3 block(128);
    hipLaunchKernelGGL(attn_fused_kernel, grid, block, 0, stream,
                       q, k, v, mask, prior, out, attn);
    (void)in_sizes; (void)n_in; (void)out_size; (void)d_ws; (void)ws_size;
}